// MNL_69260642615340
// MI455X (gfx1250) — compile-verified
//
#include <hip/hip_runtime.h>

typedef __attribute__((ext_vector_type(2))) float v2f;
typedef __attribute__((ext_vector_type(8))) float v8f;

#define NROWS 4194304
#define NFEAT 32
#define NSEG  65536

// Order-preserving float->uint encoding so unsigned atomicMax implements float max.
// f >= 0: bits | 0x80000000 ; f < 0: ~bits.  0u acts as a -inf sentinel.
__device__ __forceinline__ unsigned encodeOrderedF32(float f) {
  unsigned u = __float_as_uint(f);
  return (u & 0x80000000u) ? ~u : (u | 0x80000000u);
}
__device__ __forceinline__ float decodeOrderedF32(unsigned u) {
  return __uint_as_float((u & 0x80000000u) ? (u & 0x7FFFFFFFu) : ~u);
}

__global__ void seg_init_kernel(unsigned* __restrict__ smax, float* __restrict__ ssum) {
  int i = blockIdx.x * blockDim.x + threadIdx.x;
  if (i < NSEG) {
    smax[i] = 0u;      // decodes below any real logit
    ssum[i] = 0.0f;
  }
}

// Pass 1: logits = x . W + 2.0 via V_WMMA_F32_16X16X4_F32, 16 rows per wave.
// A (16x4 f32): lanes 0-15 = rows, VGPR0/1 = K,K+1; lanes 16-31 = K+2,K+3.
// B (4x16 f32): same K split per lane half; all 16 N-columns carry W[k], so
// every lane's C VGPR v holds logit(row r0 + v + 8*half).
__global__ __launch_bounds__(256) void logits_max_kernel(
    const float* __restrict__ x, const int* __restrict__ ids,
    const float* __restrict__ W, float* __restrict__ out,
    unsigned* __restrict__ smax) {
  const int lane = threadIdx.x & 31;
  const int wave = threadIdx.x >> 5;
  const int tile = blockIdx.x * 8 + wave;        // 16 rows per tile
  const int r0   = tile * 16;
  if (r0 >= NROWS) return;                       // wave-uniform; EXEC stays all-1s

  const int h    = lane >> 4;                    // lane half: 0 or 1
  const int m    = lane & 15;                    // A-row within tile
  const int koff = h * 2;                        // K offset for this half

  const float* xrow = x + (r0 + m) * NFEAT + koff;

  v8f c = {};
#pragma unroll
  for (int j = 0; j < 8; ++j) {                  // K = 32 in steps of 4
    v2f a = __builtin_nontemporal_load((const v2f*)(xrow + 4 * j));
    v2f b = *(const v2f*)(W + 4 * j + koff);
    c = __builtin_amdgcn_wmma_f32_16x16x4_f32(
        /*neg_a=*/false, a, /*neg_b=*/false, b,
        /*c_mod=*/(short)0, c, /*reuse_a=*/false, /*reuse_b=*/false);
  }

  // 16 writer lanes per wave: lanes 0-7 cover rows r0..r0+7 (c[v] = row r0+v),
  // lanes 16-23 cover rows r0+8..r0+15.
  if (m < 8) {
    float l = c[m] + 2.0f;
    int row = r0 + m + 8 * h;
    out[row] = l;                                // stash logits in d_out
    int s = ids[row];
    atomicMax(&smax[s], encodeOrderedF32(l));
  }
}

// Pass 2: e = exp(logit - segmax); accumulate segment sums.
__global__ __launch_bounds__(256) void exp_sum_kernel(
    const int* __restrict__ ids, const unsigned* __restrict__ smax,
    float* __restrict__ out, float* __restrict__ ssum) {
  int i = blockIdx.x * blockDim.x + threadIdx.x;
  if (i < NROWS) {
    int s = ids[i];
    float e = expf(out[i] - decodeOrderedF32(smax[s]));
    out[i] = e;
    atomicAdd(&ssum[s], e);                      // global_atomic_add_f32, L2-resident
  }
}

// Pass 3: normalize.
__global__ __launch_bounds__(256) void norm_kernel(
    const int* __restrict__ ids, const float* __restrict__ ssum,
    float* __restrict__ out) {
  int i = blockIdx.x * blockDim.x + threadIdx.x;
  if (i < NROWS) out[i] = out[i] / ssum[ids[i]];
}

extern "C" void kernel_launch(void* const* d_in, const int* in_sizes, int n_in,
                              void* d_out, int out_size, void* d_ws, size_t ws_size,
                              hipStream_t stream) {
  const float* x   = (const float*)d_in[0];   // (N, 32) f32
  const int*   ids = (const int*)  d_in[1];   // (N, 1)  i32, sorted
  const float* W   = (const float*)d_in[2];   // (1, 32) f32
  float* out = (float*)d_out;                 // N f32, also used as logit/ex scratch

  unsigned* smax = (unsigned*)d_ws;                         // S uints
  float*    ssum = (float*)((char*)d_ws + NSEG * sizeof(unsigned)); // S floats

  seg_init_kernel<<<NSEG / 256, 256, 0, stream>>>(smax, ssum);
  logits_max_kernel<<<NROWS / 128, 256, 0, stream>>>(x, ids, W, out, smax);
  exp_sum_kernel<<<NROWS / 256, 256, 0, stream>>>(ids, smax, out, ssum);
  norm_kernel<<<NROWS / 256, 256, 0, stream>>>(ids, ssum, out);
}